// AngularSymmetry_68315749810414
// MI455X (gfx1250) — compile-verified
//
#include <hip/hip_runtime.h>
#include <math.h>

typedef __attribute__((ext_vector_type(2))) float v2f;
typedef __attribute__((ext_vector_type(8))) float v8f;

#define BB 16
#define MM 64
#define PP 6

// One wave32 handles one (b,i): 4x4 tiles of 16x16 Gram matrix via
// V_WMMA_F32_16X16X4_F32 (K=4, 3 coord dims + zero pad), then the
// transcendental chain + reduction over all 4096 (j,k) pairs.
__global__ __launch_bounds__(256) void AngularSymmetry_kernel(
    const float* __restrict__ fc,     // (B,M,M) d_cutoff
    const float* __restrict__ dmat,   // (B,M,M) d
    const float* __restrict__ coords, // (B,M,3)
    float* __restrict__ out)          // (B,M,6)
{
    const int wave = threadIdx.x >> 5;           // 0..7
    const int bi   = blockIdx.x * 8 + wave;      // 0..1023
    const int b    = bi >> 6;
    const int i    = bi & 63;
    const int lane = threadIdx.x & 31;
    const int lm   = lane & 15;
    const int half = lane >> 4;                  // 0: K={0,1}, 1: K={2,3}

    const float* cb = coords + (size_t)b * MM * 3;
    const float cix = cb[i * 3 + 0];
    const float ciy = cb[i * 3 + 1];
    const float ciz = cb[i * 3 + 2];

    // Fragment for each 16-row tile. A(16x4, M on lanes%16, K split by lane
    // half across vgpr pair) and B(4x16, N on lanes%16, K split the same way)
    // have identical register contents for the Gram matrix (B = A^T data).
    // K=3 is the zero pad.
    v2f frag[4];
#pragma unroll
    for (int t = 0; t < 4; ++t) {
        const int r = t * 16 + lm;
        const float vx = cix - cb[r * 3 + 0];
        const float vy = ciy - cb[r * 3 + 1];
        const float vz = ciz - cb[r * 3 + 2];
        frag[t].x = half ? vz : vx;
        frag[t].y = half ? 0.0f : vy;
    }

    const float* db  = dmat + (size_t)b * MM * MM;
    const float* fb  = fc   + (size_t)b * MM * MM;
    const float* dri = db + (size_t)i * MM;   // R_i*  row
    const float* fri = fb + (size_t)i * MM;   // fc_i* row

    float acc0 = 0.f, acc1 = 0.f, acc2 = 0.f;   // lambd = +1, zeta = 2,4,8
    float acc3 = 0.f, acc4 = 0.f, acc5 = 0.f;   // lambd = -1, zeta = 2,4,8

#pragma unroll
    for (int kt = 0; kt < 4; ++kt) {
        const int k   = kt * 16 + lm;
        const float Rik = dri[k];
        const float Fik = fri[k];
#pragma unroll
        for (int jt = 0; jt < 4; ++jt) {
            v8f D = {};
            // D = frag[jt] (A: rows j) x frag[kt] (B: cols k) + 0
            D = __builtin_amdgcn_wmma_f32_16x16x4_f32(
                    false, frag[jt], false, frag[kt],
                    (short)0, D, false, false);
#pragma unroll
            for (int v = 0; v < 8; ++v) {
                const int j    = jt * 16 + v + half * 8;   // C/D row mapping
                const float dotv = D[v];
                const float Rij  = dri[j];
                const float Fij  = fri[j];
                const float Rjk  = db[j * MM + k];
                const float Fjk  = fb[j * MM + k];

                const float ratio = dotv / (Rij * Rik + 1e-5f);
                const float c  = cosf(ratio);
                const float ss = Rij * Rij + Rik * Rik + Rjk * Rjk;
                const float w  = expf(-4.0f * ss) * (Fij * Fik * Fjk);

                const float tp = 1.0f + c, tm = 1.0f - c;
                const float tp2 = tp * tp, tp4 = tp2 * tp2, tp8 = tp4 * tp4;
                const float tm2 = tm * tm, tm4 = tm2 * tm2, tm8 = tm4 * tm4;
                acc0 = fmaf(tp2, w, acc0);
                acc1 = fmaf(tp4, w, acc1);
                acc2 = fmaf(tp8, w, acc2);
                acc3 = fmaf(tm2, w, acc3);
                acc4 = fmaf(tm4, w, acc4);
                acc5 = fmaf(tm8, w, acc5);
            }
        }
    }

    // Cross-lane reduction over the wave32.
#pragma unroll
    for (int off = 16; off > 0; off >>= 1) {
        acc0 += __shfl_xor(acc0, off, 32);
        acc1 += __shfl_xor(acc1, off, 32);
        acc2 += __shfl_xor(acc2, off, 32);
        acc3 += __shfl_xor(acc3, off, 32);
        acc4 += __shfl_xor(acc4, off, 32);
        acc5 += __shfl_xor(acc5, off, 32);
    }

    if (lane == 0) {
        float* o = out + (size_t)bi * PP;
        o[0] = acc0 * 0.5f;        // 2^(1-2)
        o[1] = acc1 * 0.125f;      // 2^(1-4)
        o[2] = acc2 * 0.0078125f;  // 2^(1-8)
        o[3] = acc3 * 0.5f;
        o[4] = acc4 * 0.125f;
        o[5] = acc5 * 0.0078125f;
    }
}

extern "C" void kernel_launch(void* const* d_in, const int* in_sizes, int n_in,
                              void* d_out, int out_size, void* d_ws, size_t ws_size,
                              hipStream_t stream) {
    const float* fc     = (const float*)d_in[0];  // d_cutoff (16,64,64)
    const float* dmat   = (const float*)d_in[1];  // d        (16,64,64)
    const float* coords = (const float*)d_in[2];  // atom_coordinates (16,64,3)
    float* out = (float*)d_out;                   // (16,64,6)

    dim3 grid((BB * MM) / 8);   // 128 blocks
    dim3 block(256);            // 8 wave32 per block, one wave per (b,i)
    AngularSymmetry_kernel<<<grid, block, 0, stream>>>(fc, dmat, coords, out);
}